// SinusoidalAnalyzer_40269613367354
// MI455X (gfx1250) — compile-verified
//
#include <hip/hip_runtime.h>
#include <math.h>
#include <stdint.h>

// CZT sinusoidal analysis, direct-evaluated as per-frame complex GEMMs on the
// CDNA5 FP32 WMMA path (V_WMMA_F32_16X16X4_F32).
//
// yn[k] = sum_j xw[j] * exp(-i*theta*j*(k+1)),  k = u + 16*v
//       = sum_j [xw[j]*e^{-i*16*theta*j*v}] * [e^{-i*theta*j*(u+1)}]
// => C[v,u] (16 harmonics x 8 groups = 128) = A(8xK complex) * B(Kx16 complex)
// Re(A) in M rows 0..7, Im(A) in rows 8..15 -> one 16xKx16 WMMA against B_re
// and one against B_im give all four real products per K-chunk.
//
// Phases are tracked as 32-bit fixed-point fractions of a cycle:
//   phase(j) = frac(rate*j)  ==  (uint32)(pfx * j) * 2^-32   (overflow == mod 1)
// Error <= 2^-33 * j_max ~ 3e-6 rad — at f32 trig ulp level, no f64 in the loop.

typedef __attribute__((ext_vector_type(2))) float v2f;
typedef __attribute__((ext_vector_type(8))) float v8f;

#define SR_F      44100.0
#define HOP       512
#define MAX_NHAR  128
#define F0_MIN_F  40.0
#define N_FRAMES  1024
#define TWO_PI_F  6.283185307179586f
#define ANG_F     1.4629180792671596e-9f   // 2*pi / 2^32
#define POW32     4294967296.0
#define NWAVES    8

__device__ __forceinline__ uint32_t fix32(double rate) {
    // round(frac(rate) * 2^32) mod 2^32
    double fr = rate - floor(rate);
    return (uint32_t)(uint64_t)llrint(fr * POW32);
}

__global__ __launch_bounds__(32 * NWAVES)
void czt_wmma_kernel(const float* __restrict__ x, const float* __restrict__ f0,
                     float* __restrict__ out, int T)
{
    const int f    = blockIdx.x;      // frame
    const int tid  = threadIdx.x;     // 0..255
    const int lane = tid & 31;

    __shared__ float red1[256];       // [vgpr r 0..7][lane 0..31]  (A x B_re)
    __shared__ float red2[256];       //                            (A x B_im)
    red1[tid] = 0.f;
    red2[tid] = 0.f;
    __syncthreads();

    // ---- per-frame parameters (computed redundantly, all uniform) ----
    double f0v = (double)f0[f];
    if (f0v < F0_MIN_F) f0v = F0_MIN_F;
    const double delta = f0v / SR_F;                 // cycles per sample
    const int    w     = 2 * (int)rint(2.0 * SR_F / f0v);   // even winsize
    int nhar = (int)floor(SR_F / f0v * 0.5);
    if (nhar > MAX_NHAR) nhar = MAX_NHAR;

    const int xbase = f * HOP - w / 2;               // x index of j = 0

    // ---- lane roles per ISA 16x4 f32 A layout ----
    // lanes 0-15: M=0..15, components .x->K=0, .y->K=1
    // lanes 16-31: M=0..15, components .x->K=2, .y->K=3  (B symmetric)
    const int    mu    = lane & 15;                  // A row m == B col u
    const int    kbase = (lane >> 4) * 2;            // j sub-index within chunk
    const int    v     = mu & 7;
    const bool   isIm  = (mu >> 3) != 0;             // rows 8..15 carry Im(A)

    // fixed-point phase rates (cycles/sample in 2^-32 units)
    const uint32_t pfxA = fix32(16.0 * (double)v * delta);
    const uint32_t pfxB = fix32((double)(mu + 1) * delta);
    const uint32_t pfxW = fix32(1.0 / (double)w);

    v8f acc1 = {0.f, 0.f, 0.f, 0.f, 0.f, 0.f, 0.f, 0.f};
    v8f acc2 = {0.f, 0.f, 0.f, 0.f, 0.f, 0.f, 0.f, 0.f};

    // chunk count rounded so every wave runs the same number of unrolled
    // iterations; overshoot chunks have A == 0 and are harmless.
    const int wave = tid >> 5;
    const int nch  = (w + 3) >> 2;                   // K-chunks of 4 samples
    const int nch2 = (nch + 2 * NWAVES - 1) & ~(2 * NWAVES - 1);

    for (int c = wave; c < nch2; c += 2 * NWAVES) {
        #pragma unroll
        for (int half = 0; half < 2; ++half) {
            const int jb = (c + half * NWAVES) * 4 + kbase;
            v2f Aop, Bre, Bim;
            #pragma unroll
            for (int s = 0; s < 2; ++s) {
                const int      j  = jb + s;
                const uint32_t ju = (uint32_t)j;
                const int      xi = xbase + j;
                // branchless masked load: clamp index, multiply by 0/1 mask
                const int   xic = min(max(xi, 0), T - 1);
                const float m   = (xi >= 0 && xi < T && j < w) ? 1.f : 0.f;
                const float xv  = x[xic] * m;
                // periodic Blackman window via fixed-point phase
                const float aw = (float)(pfxW * ju) * ANG_F;
                const float c1 = __cosf(aw);
                const float c2 = 2.f * c1 * c1 - 1.f;          // cos(4*pi*r)
                const float xw = xv * (0.42f - 0.5f * c1 + 0.08f * c2);
                // A element: xw * e^{-i*2pi*phaseA}
                const float aa = (float)(pfxA * ju) * ANG_F;
                float sa, ca;  __sincosf(aa, &sa, &ca);
                const float Aval = isIm ? (-xw * sa) : (xw * ca);
                // B element: e^{-i*2pi*phaseB}
                const float ab = (float)(pfxB * ju) * ANG_F;
                float sb, cb;  __sincosf(ab, &sb, &cb);
                if (s == 0) { Aop.x = Aval; Bre.x = cb; Bim.x = -sb; }
                else        { Aop.y = Aval; Bre.y = cb; Bim.y = -sb; }
            }
            // 16x16x4 FP32 WMMA: rows 0-7 Re(A)*B, rows 8-15 Im(A)*B
            acc1 = __builtin_amdgcn_wmma_f32_16x16x4_f32(
                       false, Aop, false, Bre, (short)0, acc1, false, false);
            acc2 = __builtin_amdgcn_wmma_f32_16x16x4_f32(
                       false, Aop, false, Bim, (short)0, acc2, false, false);
        }
    }

    // ---- cross-wave reduction (ds_add_f32) ----
    #pragma unroll
    for (int r = 0; r < 8; ++r) {
        atomicAdd(&red1[r * 32 + lane], acc1[r]);
        atomicAdd(&red2[r * 32 + lane], acc2[r]);
    }
    __syncthreads();

    // ---- finalize: one thread per harmonic ----
    if (tid < MAX_NHAR) {
        const int h  = tid;
        const int u  = h & 15;
        const int hv = h >> 4;
        // C/D layout: VGPR r, lanes 0-15 -> M=r, lanes 16-31 -> M=r+8
        const float Xrr = red1[hv * 32 + u];        // Re(A)*Re(B)
        const float Xir = red1[hv * 32 + 16 + u];   // Im(A)*Re(B)
        const float Xri = red2[hv * 32 + u];        // Re(A)*Im(B)
        const float Xii = red2[hv * 32 + 16 + u];   // Im(A)*Im(B)
        const float sc  = 2.381f / (float)(w / 2 + 1);
        const float re  = (Xrr - Xii) * sc;
        const float im  = (Xri + Xir) * sc;
        float amp = 0.f, ph = 0.f;
        if (h < nhar) {
            amp = sqrtf(re * re + im * im);
            ph  = atan2f(im, re);
        }
        out[h * N_FRAMES + f]                       = amp;  // ampl  (128, F)
        out[MAX_NHAR * N_FRAMES + h * N_FRAMES + f] = ph;   // phase (128, F)
    }
}

extern "C" void kernel_launch(void* const* d_in, const int* in_sizes, int n_in,
                              void* d_out, int out_size, void* d_ws, size_t ws_size,
                              hipStream_t stream) {
    const float* x  = (const float*)d_in[0];   // (T,) float32
    const float* f0 = (const float*)d_in[1];   // (N_FRAMES,) float32
    float* out = (float*)d_out;                // ampl(128,F) ++ phase(128,F)
    const int T = in_sizes[0];
    czt_wmma_kernel<<<N_FRAMES, 32 * NWAVES, 0, stream>>>(x, f0, out, T);
}